// CapsuleLayer_6622839570629
// MI455X (gfx1250) — compile-verified
//
#include <hip/hip_runtime.h>
#include <math.h>

typedef __attribute__((ext_vector_type(2))) float v2f;
typedef __attribute__((ext_vector_type(8))) float v8f;

namespace caps {
constexpr int B = 256;    // batch
constexpr int I = 1152;   // input caps
constexpr int O = 10;     // output caps
constexpr int D = 16;     // out_dim
constexpr int K = 8;      // in_dim
constexpr int BT = 16;    // batch tile (WMMA M)
constexpr int ICHUNK = 72;            // input caps per block
constexpr int NCHUNK = I / ICHUNK;    // 16
constexpr int WAVES = 4;
constexpr int THREADS = WAVES * 32;   // 128
constexpr int IPW = ICHUNK / WAVES;   // 18 i's per wave
constexpr int ROUTING_ITERS = 3;
}
using namespace caps;

// D = A(16x4 f32) * B(4x16 f32) + C(16x16 f32), wave32.
// A layout: lane L holds row M=L&15; vgpr0 = K=(L>=16?2:0), vgpr1 = K+1.
// B layout: lane L holds col N=L&15; vgpr0 = K=(L>=16?2:0), vgpr1 = K+1.
// C/D layout: vgpr j -> row M = j + 8*(L>=16), col N = L&15.
__device__ __forceinline__ v8f wmma_16x16x4_f32(v2f a, v2f b, v8f c) {
  return __builtin_amdgcn_wmma_f32_16x16x4_f32(false, a, false, b,
                                               (short)0, c, false, false);
}

// s-pass: spart[chunk][b][o][d] = sum_{i in chunk} c[b,i,o] * (W[i,o] @ u[b,i])
// c folded into the A operand (per-lane row scaling) -> pure chained WMMA.
template <bool UNIFORM>
__global__ __launch_bounds__(THREADS)
void spass_kernel(const float* __restrict__ u, const float* __restrict__ W,
                  const float* __restrict__ blog, float* __restrict__ spart) {
  __shared__ float s_lds[WAVES][BT][O][D];   // 40 KB

  const int lane  = threadIdx.x & 31;
  const int wave  = threadIdx.x >> 5;
  const int b0    = blockIdx.x * BT;
  const int ibase = blockIdx.y * ICHUNK + wave * IPW;
  const int row   = lane & 15;   // batch row for A, d-column for B/C
  const int khalf = lane >> 4;   // 0 or 1
  const int k0    = khalf * 2;   // K element base for this lane

  v8f sacc[O];
  const v8f vzero = {0.f, 0.f, 0.f, 0.f, 0.f, 0.f, 0.f, 0.f};
#pragma unroll
  for (int o = 0; o < O; ++o) sacc[o] = vzero;

  for (int ii = 0; ii < IPW; ++ii) {
    const int i = ibase + ii;

    // A fragments: u[b0+row, i, k]  (K=8 -> two K=4 fragments)
    const float* up = u + ((size_t)(b0 + row) * I + i) * K;
    const v2f a0 = {up[k0], up[k0 + 1]};
    const v2f a1 = {up[4 + k0], up[4 + k0 + 1]};

    // routing coefficients for this lane's batch row (all lanes compute;
    // lanes 16..31 duplicate rows 0..15 -> no EXEC divergence around WMMA)
    float myc[O];
    if (UNIFORM) {
#pragma unroll
      for (int o = 0; o < O; ++o) myc[o] = 1.0f / (float)O;
    } else {
      const float* bp = blog + ((size_t)(b0 + row) * I + i) * O;
      float bl[O];
#pragma unroll
      for (int o = 0; o < O; ++o) bl[o] = bp[o];
      float m = bl[0];
#pragma unroll
      for (int o = 1; o < O; ++o) m = fmaxf(m, bl[o]);
      float sum = 0.f;
#pragma unroll
      for (int o = 0; o < O; ++o) { myc[o] = __expf(bl[o] - m); sum += myc[o]; }
      const float inv = 1.0f / sum;
#pragma unroll
      for (int o = 0; o < O; ++o) myc[o] *= inv;
    }

    // B fragments: B[k][d] = W[i,o,d,k]; d = row, k = k0/k0+1 (+4)
    const float* wbase = W + ((size_t)i * O * D + row) * K;
#pragma unroll
    for (int o = 0; o < O; ++o) {
      const float* wp = wbase + (size_t)o * D * K;
      const v2f bv0 = {wp[k0], wp[k0 + 1]};
      const v2f bv1 = {wp[4 + k0], wp[4 + k0 + 1]};
      const float cv = myc[o];
      sacc[o] = wmma_16x16x4_f32(a0 * cv, bv0, sacc[o]);
      sacc[o] = wmma_16x16x4_f32(a1 * cv, bv1, sacc[o]);
    }
  }

  // stage per-wave partials: accumulator vgpr j -> batch row j + 8*khalf, d=row
#pragma unroll
  for (int o = 0; o < O; ++o)
#pragma unroll
    for (int j = 0; j < 8; ++j)
      s_lds[wave][j + khalf * 8][o][row] = sacc[o][j];
  __syncthreads();

  // deterministic 4-wave reduce, write this chunk's partial (no atomics)
  const float* p = &s_lds[0][0][0][0];
  constexpr int TILE = BT * O * D;  // 2560, divisible by THREADS
  for (int e = threadIdx.x; e < TILE; e += THREADS) {
    float acc = p[e] + p[e + TILE] + p[e + 2 * TILE] + p[e + 3 * TILE];
    const int r = e / (O * D);
    const int rem = e - r * (O * D);
    spart[((size_t)blockIdx.y * B + (b0 + r)) * (O * D) + rem] = acc;
  }
}

// squash: v[b,o,:] = squash(sum_chunk spart[chunk][b,o,:])
__global__ __launch_bounds__(128)
void squash_kernel(const float* __restrict__ spart, float* __restrict__ vout) {
  const int idx = blockIdx.x * blockDim.x + threadIdx.x;  // b*O + o
  if (idx >= B * O) return;
  float sv[D];
#pragma unroll
  for (int d = 0; d < D; ++d) sv[d] = 0.f;
  for (int c = 0; c < NCHUNK; ++c) {
    const float* sp = spart + ((size_t)c * B * O + idx) * D;
#pragma unroll
    for (int d = 0; d < D; ++d) sv[d] += sp[d];
  }
  float sq = 0.f;
#pragma unroll
  for (int d = 0; d < D; ++d) sq += sv[d] * sv[d];
  const float scale = (sq / (1.0f + sq)) / (sqrtf(sq) + 1e-8f);
  float* vp = vout + (size_t)idx * D;
#pragma unroll
  for (int d = 0; d < D; ++d) vp[d] = scale * sv[d];
}

// agreement: blog[b,i,o] (+)= sum_d (W[i,o]@u[b,i])[d] * v[b,o,d]
//          = sum_k u[b,i,k] * (sum_d W[i,o,d,k]*v[b,o,d])
// one block per (i,o); W[i,o] staged in LDS (read once per block from L2)
template <bool OVERWRITE>
__global__ __launch_bounds__(B)
void agreement_kernel(const float* __restrict__ u, const float* __restrict__ W,
                      const float* __restrict__ v, float* __restrict__ blog) {
  __shared__ float wsh[D * K];  // 128 floats
  const int io = blockIdx.x;
  const int i = io / O;
  const int o = io - i * O;
  if (threadIdx.x < D * K) wsh[threadIdx.x] = W[(size_t)io * D * K + threadIdx.x];
  __syncthreads();

  const int b = threadIdx.x;
  const float* vp = v + ((size_t)b * O + o) * D;
  const float* up = u + ((size_t)b * I + i) * K;

  float t[K];
#pragma unroll
  for (int k = 0; k < K; ++k) t[k] = 0.f;
#pragma unroll
  for (int d = 0; d < D; ++d) {
    const float vv = vp[d];
#pragma unroll
    for (int k = 0; k < K; ++k) t[k] += wsh[d * K + k] * vv;
  }
  float ag = 0.f;
#pragma unroll
  for (int k = 0; k < K; ++k) ag += t[k] * up[k];

  const size_t bi = ((size_t)b * I + i) * O + o;
  if (OVERWRITE) blog[bi] = ag;
  else           blog[bi] += ag;
}

extern "C" void kernel_launch(void* const* d_in, const int* in_sizes, int n_in,
                              void* d_out, int out_size, void* d_ws, size_t ws_size,
                              hipStream_t stream) {
  (void)in_sizes; (void)n_in; (void)out_size; (void)ws_size;
  const float* u = (const float*)d_in[0];   // [B, I, K]
  const float* W = (const float*)d_in[1];   // [I, O, D, K]
  float* out = (float*)d_out;               // [B, O, D]

  // workspace layout (floats): blog | spart | v   (~14.6 MB total)
  float* blog  = (float*)d_ws;                          // B*I*O
  float* spart = blog + (size_t)B * I * O;              // NCHUNK*B*O*D
  float* vws   = spart + (size_t)NCHUNK * B * O * D;    // B*O*D

  const dim3 sgrid(B / BT, NCHUNK);  // 16 x 16 blocks

  for (int it = 0; it < ROUTING_ITERS; ++it) {
    if (it == 0)
      spass_kernel<true><<<sgrid, THREADS, 0, stream>>>(u, W, blog, spart);
    else
      spass_kernel<false><<<sgrid, THREADS, 0, stream>>>(u, W, blog, spart);

    float* vdst = (it == ROUTING_ITERS - 1) ? out : vws;
    squash_kernel<<<(B * O + 127) / 128, 128, 0, stream>>>(spart, vdst);

    if (it < ROUTING_ITERS - 1) {
      if (it == 0)
        agreement_kernel<true><<<I * O, B, 0, stream>>>(u, W, vws, blog);
      else
        agreement_kernel<false><<<I * O, B, 0, stream>>>(u, W, vws, blog);
    }
  }
}